// FocalTopLoss_83854941487537
// MI455X (gfx1250) — compile-verified
//
#include <hip/hip_runtime.h>
#include <hip/hip_bf16.h>

typedef __attribute__((ext_vector_type(2))) float v2f;
typedef __attribute__((ext_vector_type(8))) float v8f;

#define THREADS 256

// One block per row: S = sum_c exp(x[row,c]) via v_wmma_f32_16x16x4_f32 with an
// all-ones B matrix (summation is commutative, so A-tile element placement is
// irrelevant). Writes log(exp(x[row,t])/S + 1e-6) to row_out[row].
__global__ __launch_bounds__(THREADS) void FocalTop_row_kernel(
    const float* __restrict__ inp, const int* __restrict__ target,
    float* __restrict__ row_out, int C) {
  const int row  = blockIdx.x;
  const int tid  = threadIdx.x;
  const int lane = tid & 31;
  const int wave = tid >> 5;
  const float* __restrict__ rp = inp + (size_t)row * (size_t)C;

  const v2f bones = {1.0f, 1.0f};
  v8f acc = {};          // 16x16 f32 accumulator (8 VGPRs)
  float pextra = 0.0f;   // non-WMMA remainder path (empty for C % 1024 == 0)

  const int C4    = C >> 2;            // float4 chunks in the row
  const int iters = C4 / THREADS;      // UNIFORM trip count -> scalar loop ctrl

  const float4* __restrict__ p = (const float4*)rp + tid;  // per-lane base
#pragma unroll 2
  for (int it = 0; it < iters; ++it) {
    const float4 x = p[it * THREADS];                      // coalesced b128
    // speculative prefetch 2 iterations ahead; OOB prefetch is silently
    // dropped (ISA 10.5). locality 3 -> WGP scope (fill all cache levels).
    __builtin_prefetch(p + (it + 2) * THREADS, 0, 3);
    v2f a0 = { __expf(x.x), __expf(x.y) };
    v2f a1 = { __expf(x.z), __expf(x.w) };
    // D = A x ones + C : folds 64 exps into the accumulator per instruction
    acc = __builtin_amdgcn_wmma_f32_16x16x4_f32(false, a0, false, bones,
                                                (short)0, acc, false, false);
    acc = __builtin_amdgcn_wmma_f32_16x16x4_f32(false, a1, false, bones,
                                                (short)0, acc, false, false);
  }
  // remainder float4 chunks (partial-wave) — scalar path, keeps WMMA EXEC all-1s
  for (int c4 = iters * THREADS + tid; c4 < C4; c4 += THREADS) {
    const float4 x = *(const float4*)(rp + (c4 << 2));
    pextra += __expf(x.x) + __expf(x.y) + __expf(x.z) + __expf(x.w);
  }
  // remainder scalars
  for (int c = (C4 << 2) + tid; c < C; c += THREADS) pextra += __expf(rp[c]);

  // Collapse accumulator: lanes 0-15 hold rows 0-7 in acc[0..7] (N-replicated),
  // lanes 16-31 hold rows 8-15. One xor-16 shuffle completes the wave sum.
  float v = acc[0] + acc[1] + acc[2] + acc[3] + acc[4] + acc[5] + acc[6] + acc[7];
  v += __shfl_xor(v, 16, 32);

  __shared__ float wsum[THREADS / 32];
  __shared__ float red[THREADS];
  red[tid] = pextra;
  if (lane == 0) wsum[wave] = v;
  __syncthreads();
  for (int s = THREADS >> 1; s > 0; s >>= 1) {
    if (tid < s) red[tid] += red[tid + s];
    __syncthreads();
  }
  if (tid == 0) {
    float S = red[0];
#pragma unroll
    for (int w = 0; w < THREADS / 32; ++w) S += wsum[w];
    const int t  = target[row];
    const float pt = __expf(rp[t]) / S;
    row_out[row] = __logf(pt + 1e-6f);
  }
}

__global__ __launch_bounds__(THREADS) void FocalTop_mean_kernel(
    const float* __restrict__ rv, float* __restrict__ out, int B) {
  __shared__ float red[THREADS];
  float s = 0.0f;
  for (int i = threadIdx.x; i < B; i += THREADS) s += rv[i];
  red[threadIdx.x] = s;
  __syncthreads();
  for (int st = THREADS >> 1; st > 0; st >>= 1) {
    if (threadIdx.x < st) red[threadIdx.x] += red[threadIdx.x + st];
    __syncthreads();
  }
  if (threadIdx.x == 0) out[0] = -red[0] / (float)B;
}

extern "C" void kernel_launch(void* const* d_in, const int* in_sizes, int n_in,
                              void* d_out, int out_size, void* d_ws, size_t ws_size,
                              hipStream_t stream) {
  const float* inp    = (const float*)d_in[0];
  const int*   target = (const int*)d_in[1];
  const int B = in_sizes[1];
  const int C = in_sizes[0] / B;
  float* row_out = (float*)d_ws;   // B floats of scratch

  FocalTop_row_kernel<<<B, THREADS, 0, stream>>>(inp, target, row_out, C);
  FocalTop_mean_kernel<<<1, THREADS, 0, stream>>>(row_out, (float*)d_out, B);
}